// TimeCorrelatedConv_7086696038823
// MI455X (gfx1250) — compile-verified
//
#include <hip/hip_runtime.h>
#include <hip/hip_bf16.h>
#include <math.h>

typedef float v2f __attribute__((ext_vector_type(2)));
typedef float v8f __attribute__((ext_vector_type(8)));

static constexpr int kB = 8;     // batch
static constexpr int kT = 128;   // T1 == T2
static constexpr int kC = 256;   // channels
static constexpr int kN = 768;   // 3 * kC  (GEMM N dimension)

// workspace layout (in floats)
static constexpr size_t WP1_OFF = 0;                       // 9*65536
static constexpr size_t WP2_OFF = WP1_OFF + 9 * 65536;     // 9*65536
static constexpr size_t M1_OFF  = WP2_OFF + 9 * 65536;     // 8*128*768
static constexpr size_t M2_OFF  = M1_OFF + (size_t)kB * kT * kN;

// ---------------------------------------------------------------------------
// Kernel 0: repack W (OIHW, [256][512][3][3]) into fragment-major B operands:
//   WpF[s][blk][kt][h][o][v]  (flat r = sb*65536 + kt*1024 + h*512 + o*2 + v)
// holding W[o][p*256 + i][kh][kw] with i = kt*4 + v + 2h and
//   part1 (p=0): kh = s (shift dim), kw = blk
//   part2 (p=1): kw = s (shift dim), kh = blk
// A lane (m,h) of B-fragment (s,blk,kt) then reads a contiguous float pair at
// ((sb*64+kt)*2 + h)*256 + o, i.e. one coalesced global_load_b64.
// ---------------------------------------------------------------------------
__global__ __launch_bounds__(256) void repack_w(const float* __restrict__ W,
                                                float* __restrict__ Wp1,
                                                float* __restrict__ Wp2) {
  int f = blockIdx.x * blockDim.x + threadIdx.x; // 0 .. 2*589824-1
  int p = f / 589824;
  int r = f % 589824;
  int v   = r & 1;
  int o   = (r >> 1) & 255;
  int h   = (r >> 9) & 1;
  int kt  = (r >> 10) & 63;
  int sb  = r >> 16;          // 0..8 = s*3 + blk
  int s   = sb / 3;
  int blk = sb % 3;
  int i = kt * 4 + v + 2 * h;
  int kh, kw;
  if (p == 0) { kh = s; kw = blk; } else { kw = s; kh = blk; }
  float val = W[(size_t)o * (512 * 9) + (size_t)(p * 256 + i) * 9 + kh * 3 + kw];
  (p == 0 ? Wp1 : Wp2)[r] = val;
}

// ---------------------------------------------------------------------------
// Kernel 1: stage-1 GEMM with V_WMMA_F32_16X16X4_F32, M-blocked x2.
//   Mout[b][t][blk*256+o] = sum_{s=0..2} sum_i Xpad[b][t+s-1][i]*W[...]
// Block = 256 threads (8 wave32), 32 output rows x 768 cols.
// A rows (32 + 2 halo) staged in LDS via GLOBAL_LOAD_ASYNC_TO_LDS_B32
// (ASYNCcnt path), stride-260 skew -> conflict-free ds_load_b64 A fragments.
// Per k-step per wave: 2 LDS b64 (A) + 6 global b64 (B) -> 12 WMMAs.
// ---------------------------------------------------------------------------
__global__ __launch_bounds__(256) void gemm_stage1(const float* __restrict__ X,
                                                   const float* __restrict__ WpF,
                                                   float* __restrict__ Mout) {
  __shared__ float lds[34 * 260];

  const int b   = blockIdx.y;
  const int t0  = blockIdx.x * 32;
  const int tid = threadIdx.x;

  if (tid == 0) __builtin_prefetch(WpF, 0, 1); // global_prefetch_b8 hint

  // Pre-zero the (at most one) OOB halo row, then async-DMA the valid rows.
  if (t0 == 0)        lds[0 * 260 + tid]  = 0.0f;  // gt = -1
  if (t0 + 32 == kT)  lds[33 * 260 + tid] = 0.0f;  // gt = 128
  for (int it = 0; it < 34; ++it) {
    int gt = t0 - 1 + it;               // uniform guard per iteration
    if (gt >= 0 && gt < kT) {
      uint32_t laddr = (uint32_t)(uintptr_t)(&lds[it * 260 + tid]); // LDS byte addr
      uint64_t gaddr = (uint64_t)(uintptr_t)(X + ((size_t)b * kT + gt) * kC + tid);
      asm volatile("global_load_async_to_lds_b32 %0, %1, off"
                   :: "v"(laddr), "v"(gaddr) : "memory");
    }
  }
  asm volatile("s_wait_asynccnt 0x0" ::: "memory");
  __syncthreads();

  const int wave = tid >> 5;
  const int lane = tid & 31;
  const int m = lane & 15;  // A-row / B-col lane index
  const int h = lane >> 4;  // K half select

  v8f acc[2][6];
#pragma unroll
  for (int mt = 0; mt < 2; ++mt)
#pragma unroll
    for (int n = 0; n < 6; ++n)
      acc[mt][n] = (v8f){0.f, 0.f, 0.f, 0.f, 0.f, 0.f, 0.f, 0.f};

  int jt[6], kwb[6], ob[6];
#pragma unroll
  for (int n = 0; n < 6; ++n) {
    jt[n]  = wave + 8 * n;      // N-tile id, 0..47
    kwb[n] = jt[n] >> 4;        // which 256-wide (kw/kh) block
    ob[n]  = (jt[n] & 15) * 16; // o base within block
  }

  for (int s = 0; s < 3; ++s) {
    for (int kt = 0; kt < 64; ++kt) {
      int i0 = kt * 4;
      // A fragments (docs 7.12.2): A[m][k=v+2h] = Xpad[t0+mt*16+m+s-1][i0+v+2h]
      v2f a0 = *(const v2f*)&lds[(m + s) * 260 + i0 + 2 * h];
      v2f a1 = *(const v2f*)&lds[(m + 16 + s) * 260 + i0 + 2 * h];
#pragma unroll
      for (int n = 0; n < 6; ++n) {
        // B fragment: one contiguous float pair per lane (global_load_b64)
        const float* bp =
            WpF + ((((size_t)(s * 3 + kwb[n]) * 64 + kt) * 2 + h) * 256 + ob[n] + m) * 2;
        v2f bf = *(const v2f*)bp;
        acc[0][n] = __builtin_amdgcn_wmma_f32_16x16x4_f32(
            false, a0, false, bf, (short)0, acc[0][n], false, false);
        acc[1][n] = __builtin_amdgcn_wmma_f32_16x16x4_f32(
            false, a1, false, bf, (short)0, acc[1][n], false, false);
      }
    }
  }

  // C/D layout: VGPR r -> row (r + 8h), col = m   (docs 7.12.2)
#pragma unroll
  for (int mt = 0; mt < 2; ++mt)
#pragma unroll
    for (int n = 0; n < 6; ++n)
#pragma unroll
      for (int r = 0; r < 8; ++r) {
        int row = t0 + 16 * mt + r + 8 * h;
        Mout[((size_t)b * kT + row) * kN + 16 * jt[n] + m] = acc[mt][n][r];
      }
}

// ---------------------------------------------------------------------------
// Kernel 2: classes + bias + exact GELU + mean over t2 + alpha*... + x1.
//   P1 classes over kw (t2 edge): L = kw1+kw2, M = all, R = kw0+kw1
//   P2 classes over kh (t1 edge): weights (w0,1,w2), w0=0 iff t1==0,
//                                 w2=0 iff t1==T-1
// Block per (b,t1); thread per o; loop over t2 with coalesced loads.
// ---------------------------------------------------------------------------
__global__ __launch_bounds__(256) void finalize_k(const float* __restrict__ x1,
                                                  const float* __restrict__ M1,
                                                  const float* __restrict__ M2,
                                                  const float* __restrict__ bias,
                                                  const float* __restrict__ alpha_p,
                                                  float* __restrict__ out) {
  const int b  = blockIdx.x >> 7;
  const int t1 = blockIdx.x & 127;
  const int o  = threadIdx.x;

  const float* m1r = M1 + ((size_t)b * kT + t1) * kN;
  float m1_0 = m1r[o], m1_1 = m1r[kC + o], m1_2 = m1r[2 * kC + o];
  float P1M = m1_0 + m1_1 + m1_2;
  float P1L = m1_1 + m1_2;   // t2 == 0   -> kw=0 falls outside
  float P1R = m1_0 + m1_1;   // t2 == T-1 -> kw=2 falls outside

  float w0 = (t1 == 0)      ? 0.0f : 1.0f;  // kh=0 invalid at t1==0
  float w2 = (t1 == kT - 1) ? 0.0f : 1.0f;  // kh=2 invalid at t1==T-1
  float bv = bias[o];

  float accv = 0.0f;
  const float* m2b = M2 + (size_t)b * kT * kN;
  for (int t2 = 0; t2 < kT; ++t2) {
    const float* m2r = m2b + (size_t)t2 * kN;
    float p2 = w0 * m2r[o] + m2r[kC + o] + w2 * m2r[2 * kC + o];
    float p1 = (t2 == 0) ? P1L : ((t2 == kT - 1) ? P1R : P1M);
    float sx = p1 + p2 + bv;
    accv += 0.5f * sx * (1.0f + erff(sx * 0.70710678118654752f)); // exact GELU
  }

  size_t idx = ((size_t)b * kT + t1) * kC + o;
  out[idx] = x1[idx] + alpha_p[0] * accv * (1.0f / 128.0f);
}

// ---------------------------------------------------------------------------
extern "C" void kernel_launch(void* const* d_in, const int* in_sizes, int n_in,
                              void* d_out, int out_size, void* d_ws, size_t ws_size,
                              hipStream_t stream) {
  const float* x1    = (const float*)d_in[0];
  const float* x2    = (const float*)d_in[1];
  const float* W     = (const float*)d_in[2];
  const float* bias  = (const float*)d_in[3];
  const float* alpha = (const float*)d_in[4];

  float* ws  = (float*)d_ws;
  float* Wp1 = ws + WP1_OFF;
  float* Wp2 = ws + WP2_OFF;
  float* M1  = ws + M1_OFF;
  float* M2  = ws + M2_OFF;

  // 1) repack weights into fragment-major layout (2 * 9*65536 elements)
  repack_w<<<(2 * 9 * 65536) / 256, 256, 0, stream>>>(W, Wp1, Wp2);

  // 2) two f32-WMMA GEMMs: M1 from x1/Wp1, M2 from x2/Wp2
  gemm_stage1<<<dim3(kT / 32, kB), 256, 0, stream>>>(x1, Wp1, M1);
  gemm_stage1<<<dim3(kT / 32, kB), 256, 0, stream>>>(x2, Wp2, M2);

  // 3) GELU + mean + residual
  finalize_k<<<kB * kT, 256, 0, stream>>>(x1, M1, M2, bias, alpha, (float*)d_out);
}